// GADBase_12094627906043
// MI455X (gfx1250) — compile-verified
//
#include <hip/hip_runtime.h>
#include <math.h>

typedef float v2f __attribute__((ext_vector_type(2)));
typedef float v8f __attribute__((ext_vector_type(8)));

#define HH 1024
#define WW 1024
#define PLANE 1048576            // 1024*1024
#define NTOT  4194304            // 4 * PLANE
#define LP 33                    // padded LDS row stride (bank-conflict free)
#define TWO_PI 6.28318530717958647692f

static __device__ __forceinline__ v8f wmma4(v2f a, v2f b, v8f c) {
  // D(16x16,f32) = A(16x4,f32) * B(4x16,f32) + C  -> v_wmma_f32_16x16x4_f32
  return __builtin_amdgcn_wmma_f32_16x16x4_f32(false, a, false, b, (short)0, c,
                                               false, false);
}

// ---- CDNA5 async global->LDS copy (ASYNCcnt path), compile-safe guards ----
#if __has_builtin(__builtin_amdgcn_global_load_async_to_lds_b32)
#define HAVE_ASYNC_COPY 1
static __device__ __forceinline__ void async_g2l_b32(const float* g, float* l) {
  __builtin_amdgcn_global_load_async_to_lds_b32(
      (__attribute__((address_space(1))) int*)g,
      (__attribute__((address_space(3))) int*)l, 0, 0);
}
static __device__ __forceinline__ void wait_asynccnt0() {
#if __has_builtin(__builtin_amdgcn_s_wait_asynccnt)
  __builtin_amdgcn_s_wait_asynccnt(0);
#else
  asm volatile("s_wait_asynccnt 0" ::: "memory");
#endif
}
#endif

static __device__ __forceinline__ int refl1024(int x) {
  if (x < 0) x = -x;
  if (x >= HH) x = 2 * HH - 2 - x;
  return x;
}

// ---------------------------------------------------------------------------
// 1024-point complex FFT per line via 32x32 Cooley-Tukey, WMMA f32 matmuls.
// One wave per line, 4 lines per 128-thread block. dir_sign=-1 fwd, +1 inv.
// mode: 0 = plain, 1 = multiply HF radial mask at store (freq layout),
//       2 = 3x3 reflect-padded blur fused at load (freq layout).
// ---------------------------------------------------------------------------
__global__ __launch_bounds__(128) void fft1024_lines(
    const float* __restrict__ in_re, const float* __restrict__ in_im,
    float* __restrict__ out_re, float* __restrict__ out_im,
    long in_plane_stride, long out_plane_stride,
    float dir_sign, float scale, int mode)
{
  __shared__ float sFr [32 * LP];
  __shared__ float sFi [32 * LP];
  __shared__ float sNFi[32 * LP];
  __shared__ float sTwC[32 * LP];
  __shared__ float sTwS[32 * LP];
  __shared__ float sXr[4][32 * LP];
  __shared__ float sXi[4][32 * LP];

  const int tid  = threadIdx.x;
  const int lane = tid & 31;
  const int wave = tid >> 5;

  // DFT factor F[j][k] = exp(dir*i*2pi*j*k/32) and twiddle LUT
  // Tw[j][k] = exp(dir*i*2pi*j*k/1024). Reduce products mod period first.
  for (int t = tid; t < 1024; t += 128) {
    int j = t >> 5, k = t & 31;
    int m32 = (j * k) & 31;
    float angF = dir_sign * TWO_PI * (float)m32 * (1.0f / 32.0f);
    float s, c;
    sincosf(angF, &s, &c);
    sFr [j * LP + k] = c;
    sFi [j * LP + k] = s;
    sNFi[j * LP + k] = -s;
    int m1024 = (j * k) & 1023;
    float angT = dir_sign * TWO_PI * (float)m1024 * (1.0f / 1024.0f);
    sincosf(angT, &s, &c);
    sTwC[j * LP + k] = c;
    sTwS[j * LP + k] = s;
  }
  __syncthreads();

  const int line = blockIdx.x * 4 + wave;        // grid = 1024 -> 4096 lines
  const int p = line >> 10;
  const int r = line & 1023;
  const float* src_re = in_re + (long)p * in_plane_stride + (long)r * WW;
  const float* src_im = in_im ? (in_im + (long)p * in_plane_stride + (long)r * WW)
                              : (const float*)nullptr;
  float* dst_re = out_re + (long)p * out_plane_stride + (long)r * WW;
  float* dst_im = out_im + (long)p * out_plane_stride + (long)r * WW;

  float* Xr = sXr[wave];
  float* Xi = sXi[wave];

  // ---- Load line -> X[n1][n2] ----
  if (mode == 2) {
    // fused 3x3 blur (taps at x=-2,-0.5,1, sigma=0.1), reflect padding
    float e0 = expf(-200.0f), e1 = expf(-12.5f), e2 = expf(-50.0f);
    float inv = 1.0f / (e0 + e1 + e2);
    float g0 = e0 * inv, g1 = e1 * inv, g2 = e2 * inv;
    const int rm = refl1024(r - 1), rp = refl1024(r + 1);
    const float* r0 = in_re + (long)p * in_plane_stride + (long)rm * WW;
    const float* r1 = src_re;
    const float* r2 = in_re + (long)p * in_plane_stride + (long)rp * WW;
    const float* q0 = in_im + (long)p * in_plane_stride + (long)rm * WW;
    const float* q1 = src_im;
    const float* q2 = in_im + (long)p * in_plane_stride + (long)rp * WW;
    for (int t = lane; t < 1024; t += 32) {
      int cm = refl1024(t - 1), cp = refl1024(t + 1);
      float vr = g0 * (g0 * r0[cm] + g1 * r0[t] + g2 * r0[cp]) +
                 g1 * (g0 * r1[cm] + g1 * r1[t] + g2 * r1[cp]) +
                 g2 * (g0 * r2[cm] + g1 * r2[t] + g2 * r2[cp]);
      float vi = g0 * (g0 * q0[cm] + g1 * q0[t] + g2 * q0[cp]) +
                 g1 * (g0 * q1[cm] + g1 * q1[t] + g2 * q1[cp]) +
                 g2 * (g0 * q2[cm] + g1 * q2[t] + g2 * q2[cp]);
      int n1 = t >> 5, n2 = t & 31;
      Xr[n1 * LP + n2] = vr;
      Xi[n1 * LP + n2] = vi;
    }
  } else {
#ifdef HAVE_ASYNC_COPY
    for (int t = lane; t < 1024; t += 32) {
      int n1 = t >> 5, n2 = t & 31;
      async_g2l_b32(src_re + t, &Xr[n1 * LP + n2]);
      if (src_im) async_g2l_b32(src_im + t, &Xi[n1 * LP + n2]);
      else        Xi[n1 * LP + n2] = 0.0f;
    }
    wait_asynccnt0();
#else
    for (int t4 = lane * 4; t4 < 1024; t4 += 128) {
      __builtin_prefetch((const void*)(src_re + t4 + 512), 0, 0);
      const int n1 = t4 >> 5, n2 = t4 & 31;
      float4 v = *(const float4*)(src_re + t4);
      Xr[n1 * LP + n2 + 0] = v.x;
      Xr[n1 * LP + n2 + 1] = v.y;
      Xr[n1 * LP + n2 + 2] = v.z;
      Xr[n1 * LP + n2 + 3] = v.w;
      if (src_im) {
        float4 w = *(const float4*)(src_im + t4);
        Xi[n1 * LP + n2 + 0] = w.x;
        Xi[n1 * LP + n2 + 1] = w.y;
        Xi[n1 * LP + n2 + 2] = w.z;
        Xi[n1 * LP + n2 + 3] = w.w;
      } else {
        Xi[n1 * LP + n2 + 0] = 0.0f;
        Xi[n1 * LP + n2 + 1] = 0.0f;
        Xi[n1 * LP + n2 + 2] = 0.0f;
        Xi[n1 * LP + n2 + 3] = 0.0f;
      }
    }
#endif
  }

  const int mlane  = lane & 15;
  const int khalf  = (lane >> 4) << 1;  // 0 or 2 (A/B fragment K split)
  const int mhalf8 = (lane >> 4) << 3;  // 0 or 8 (D fragment M split)

  // ---- Step 1: P = F * X  (A = F, B = X) ----
  v8f pR[2][2], pI[2][2];
#pragma unroll
  for (int ti = 0; ti < 2; ++ti) {
#pragma unroll
    for (int tj = 0; tj < 2; ++tj) {
      const int mo = ti * 16, no = tj * 16;
      v8f aR = {0, 0, 0, 0, 0, 0, 0, 0};
      v8f aI = {0, 0, 0, 0, 0, 0, 0, 0};
      const int am = mo + mlane;
      const int bn = no + mlane;
#pragma unroll
      for (int kk = 0; kk < 8; ++kk) {
        const int ko = kk * 4;
        const int ak = ko + khalf;
        v2f fA_r  = { sFr [am * LP + ak], sFr [am * LP + ak + 1] };
        v2f fA_i  = { sFi [am * LP + ak], sFi [am * LP + ak + 1] };
        v2f fA_ni = { sNFi[am * LP + ak], sNFi[am * LP + ak + 1] };
        const int rb = ko + khalf;
        v2f bXr = { Xr[rb * LP + bn], Xr[(rb + 1) * LP + bn] };
        v2f bXi = { Xi[rb * LP + bn], Xi[(rb + 1) * LP + bn] };
        aR = wmma4(fA_r,  bXr, aR);
        aR = wmma4(fA_ni, bXi, aR);   // -Fi * Xi
        aI = wmma4(fA_r,  bXi, aI);
        aI = wmma4(fA_i,  bXr, aI);
      }
      pR[ti][tj] = aR;
      pI[ti][tj] = aI;
    }
  }

  // ---- Twiddle (LDS LUT) and write Z back into X LDS ----
#pragma unroll
  for (int ti = 0; ti < 2; ++ti) {
#pragma unroll
    for (int tj = 0; tj < 2; ++tj) {
      const int mo = ti * 16, no = tj * 16;
      const int n = no + mlane;
#pragma unroll
      for (int i = 0; i < 8; ++i) {
        const int m = mo + i + mhalf8;           // k1
        float c = sTwC[m * LP + n];
        float s = sTwS[m * LP + n];
        float prv = pR[ti][tj][i], piv = pI[ti][tj][i];
        Xr[m * LP + n] = prv * c - piv * s;
        Xi[m * LP + n] = prv * s + piv * c;
      }
    }
  }

  // ---- Step 3: O = Z * F  (A = Z, B = F) ----
  v8f oR[2][2], oI[2][2];
#pragma unroll
  for (int ti = 0; ti < 2; ++ti) {
#pragma unroll
    for (int tj = 0; tj < 2; ++tj) {
      const int mo = ti * 16, no = tj * 16;
      v8f aR = {0, 0, 0, 0, 0, 0, 0, 0};
      v8f aI = {0, 0, 0, 0, 0, 0, 0, 0};
      const int am = mo + mlane;
      const int bn = no + mlane;
#pragma unroll
      for (int kk = 0; kk < 8; ++kk) {
        const int ko = kk * 4;
        const int ak = ko + khalf;
        v2f aZr = { Xr[am * LP + ak], Xr[am * LP + ak + 1] };
        v2f aZi = { Xi[am * LP + ak], Xi[am * LP + ak + 1] };
        const int rb = ko + khalf;
        v2f bFr  = { sFr [rb * LP + bn], sFr [(rb + 1) * LP + bn] };
        v2f bFi  = { sFi [rb * LP + bn], sFi [(rb + 1) * LP + bn] };
        v2f bNFi = { sNFi[rb * LP + bn], sNFi[(rb + 1) * LP + bn] };
        aR = wmma4(aZr, bFr,  aR);
        aR = wmma4(aZi, bNFi, aR);
        aI = wmma4(aZr, bFi,  aI);
        aI = wmma4(aZi, bFr,  aI);
      }
      oR[ti][tj] = aR;
      oI[ti][tj] = aI;
    }
  }

  // ---- Store O[k1][k2] to LDS, then vectorized write out[32*k2+k1]*scale ----
#pragma unroll
  for (int ti = 0; ti < 2; ++ti) {
#pragma unroll
    for (int tj = 0; tj < 2; ++tj) {
      const int mo = ti * 16, no = tj * 16;
      const int n = no + mlane;
#pragma unroll
      for (int i = 0; i < 8; ++i) {
        const int m = mo + i + mhalf8;
        Xr[m * LP + n] = oR[ti][tj][i];
        Xi[m * LP + n] = oI[ti][tj][i];
      }
    }
  }
  if (mode == 1) {
    // fused radial high-pass mask: keep if sqrt(u(r)^2+u(t)^2) > 0.5
    float ur = -1.0f + 2.0f * (float)r * (1.0f / 1023.0f);
    for (int t4 = lane * 4; t4 < 1024; t4 += 128) {
      const int k2 = t4 >> 5, k1 = t4 & 31;   // 4 consecutive t share k2
      float kp[4];
#pragma unroll
      for (int e = 0; e < 4; ++e) {
        float ut = -1.0f + 2.0f * (float)(t4 + e) * (1.0f / 1023.0f);
        kp[e] = (sqrtf(ur * ur + ut * ut) > 0.5f) ? scale : 0.0f;
      }
      float4 vr, vi;
      vr.x = Xr[(k1 + 0) * LP + k2] * kp[0];
      vr.y = Xr[(k1 + 1) * LP + k2] * kp[1];
      vr.z = Xr[(k1 + 2) * LP + k2] * kp[2];
      vr.w = Xr[(k1 + 3) * LP + k2] * kp[3];
      vi.x = Xi[(k1 + 0) * LP + k2] * kp[0];
      vi.y = Xi[(k1 + 1) * LP + k2] * kp[1];
      vi.z = Xi[(k1 + 2) * LP + k2] * kp[2];
      vi.w = Xi[(k1 + 3) * LP + k2] * kp[3];
      *(float4*)(dst_re + t4) = vr;
      *(float4*)(dst_im + t4) = vi;
    }
  } else {
    for (int t4 = lane * 4; t4 < 1024; t4 += 128) {
      const int k2 = t4 >> 5, k1 = t4 & 31;
      float4 vr, vi;
      vr.x = Xr[(k1 + 0) * LP + k2] * scale;
      vr.y = Xr[(k1 + 1) * LP + k2] * scale;
      vr.z = Xr[(k1 + 2) * LP + k2] * scale;
      vr.w = Xr[(k1 + 3) * LP + k2] * scale;
      vi.x = Xi[(k1 + 0) * LP + k2] * scale;
      vi.y = Xi[(k1 + 1) * LP + k2] * scale;
      vi.z = Xi[(k1 + 2) * LP + k2] * scale;
      vi.w = Xi[(k1 + 3) * LP + k2] * scale;
      *(float4*)(dst_re + t4) = vr;
      *(float4*)(dst_im + t4) = vi;
    }
  }
}

// ---------------------------------------------------------------------------
// Pointwise / helper kernels (bandwidth bound, float4-vectorized where legal)
// ---------------------------------------------------------------------------
__global__ __launch_bounds__(256) void transpose2_kernel(
    const float* __restrict__ inA, float* __restrict__ outA,
    const float* __restrict__ inB, float* __restrict__ outB)
{
  __shared__ float t0[32][33];
  __shared__ float t1[32][33];
  const int p = blockIdx.z;
  const int x0 = blockIdx.x * 32, y0 = blockIdx.y * 32;
  const float* ipA = inA + (long)p * PLANE;
  const float* ipB = inB + (long)p * PLANE;
  float* opA = outA + (long)p * PLANE;
  float* opB = outB + (long)p * PLANE;
  const int tx = threadIdx.x & 31, ty = threadIdx.x >> 5;  // 32 x 8
#pragma unroll
  for (int dy = 0; dy < 32; dy += 8) {
    t0[ty + dy][tx] = ipA[(long)(y0 + ty + dy) * WW + (x0 + tx)];
    t1[ty + dy][tx] = ipB[(long)(y0 + ty + dy) * WW + (x0 + tx)];
  }
  __syncthreads();
#pragma unroll
  for (int dy = 0; dy < 32; dy += 8) {
    opA[(long)(x0 + ty + dy) * WW + (y0 + tx)] = t0[tx][ty + dy];
    opB[(long)(x0 + ty + dy) * WW + (y0 + tx)] = t1[tx][ty + dy];
  }
}

__global__ void zero4_kernel(float4* __restrict__ buf) {
  int idx = blockIdx.x * 256 + threadIdx.x;
  float4 z; z.x = 0.0f; z.y = 0.0f; z.z = 0.0f; z.w = 0.0f;
  buf[idx] = z;
}

__global__ void init_scalars_kernel(float* scal) {
  ((unsigned*)scal)[0] = 0x7f800000u;  // +inf bits for min reduction
  scal[1] = 0.0f;
}

__global__ void minreduce_kernel(const float* __restrict__ src, int n,
                                 unsigned* __restrict__ out_bits) {
  __shared__ float sm[256];
  const int tid = threadIdx.x;
  float v = 3.4e38f;
  int i = blockIdx.x * 1024 + tid;
#pragma unroll
  for (int k = 0; k < 4; ++k) {
    int idx = i + k * 256;
    if (idx < n) v = fminf(v, src[idx]);
  }
  sm[tid] = v;
  __syncthreads();
  for (int s = 128; s > 0; s >>= 1) {
    if (tid < s) sm[tid] = fminf(sm[tid], sm[tid + s]);
    __syncthreads();
  }
  if (tid == 0) atomicMin(out_bits, __float_as_uint(sm[0]));  // src >= 0
}

__global__ void shift_kernel(float* scal) {
  float m = __uint_as_float(((unsigned*)scal)[0]);
  scal[1] = (m <= 0.1f) ? 0.1f : 0.0f;
}

__global__ void init_img4_kernel(const float4* __restrict__ yb,
                                 const float* __restrict__ scal,
                                 float4* __restrict__ img) {
  int idx = blockIdx.x * 256 + threadIdx.x;
  float sh = scal[1];
  float4 v = yb[idx];
  v.x += sh; v.y += sh; v.z += sh; v.w += sh;
  img[idx] = v;
}

// Accumulate per-channel |grad| + edge response into dv/dh sums.
__global__ void edge_accum_kernel(const float* __restrict__ chan,
                                  long chan_plane_stride,
                                  const float* __restrict__ er,
                                  float* __restrict__ dvsum,
                                  float* __restrict__ dhsum) {
  int idx = blockIdx.x * 256 + threadIdx.x;
  int p = idx >> 20;
  int pix = idx & (PLANE - 1);
  int r = pix >> 10, c = pix & 1023;
  const float* I = chan + (long)p * chan_plane_stride;
  const float* E = er + (long)p * PLANE;
  __builtin_prefetch((const void*)(I + r * WW + c + WW), 0, 0);
  float center = I[r * WW + c];
  if (r < HH - 1)
    dvsum[idx] += fabsf(I[(r + 1) * WW + c] - center) + E[(r + 1) * WW + c];
  if (c < WW - 1)
    dhsum[idx] += fabsf(I[r * WW + c + 1] - center) + E[r * WW + c + 1];
}

// Perona-Malik conductance from the 4-channel mean.
__global__ void pm_finalize4_kernel(float4* __restrict__ buf) {
  int idx = blockIdx.x * 256 + threadIdx.x;
  const float K2I = 1.0f / (0.03f * 0.03f);
  float4 v = buf[idx];
  float a;
  a = v.x * 0.25f; v.x = 1.0f / (1.0f + a * a * K2I);
  a = v.y * 0.25f; v.y = 1.0f / (1.0f + a * a * K2I);
  a = v.z * 0.25f; v.z = 1.0f / (1.0f + a * a * K2I);
  a = v.w * 0.25f; v.w = 1.0f / (1.0f + a * a * K2I);
  buf[idx] = v;
}

__global__ void diffuse_kernel(const float* __restrict__ img,
                               const float* __restrict__ ifreq,
                               const float* __restrict__ cv,
                               const float* __restrict__ chh,
                               float* __restrict__ out) {
  int idx = blockIdx.x * 256 + threadIdx.x;
  int p = idx >> 20;
  int pix = idx & (PLANE - 1);
  int r = pix >> 10, c = pix & 1023;
  const float* I = img + (long)p * PLANE;
  float center = I[r * WW + c];
  float acc = 0.7f * center + 0.3f * ifreq[idx];
  if (r < HH - 1) acc += 0.24f * cv[idx] * (I[(r + 1) * WW + c] - center);
  if (r > 0)      acc -= 0.24f * cv[idx - WW] * (center - I[(r - 1) * WW + c]);
  if (c < WW - 1) acc += 0.24f * chh[idx] * (I[r * WW + c + 1] - center);
  if (c > 0)      acc -= 0.24f * chh[idx - 1] * (center - I[r * WW + c - 1]);
  out[idx] = acc;
}

__global__ void blockmean_kernel(const float* __restrict__ in,
                                 float* __restrict__ out) {
  int t = blockIdx.x * 256 + threadIdx.x;  // 4*128*128 = 65536
  int p = t >> 14;
  int pix = t & 16383;
  int br = pix >> 7, bc = pix & 127;
  const float* I = in + (long)p * PLANE + (long)(br * 8) * WW + bc * 8;
  float s = 0.0f;
#pragma unroll
  for (int a = 0; a < 8; ++a) {
    float4 u = *(const float4*)(I + a * WW);
    float4 v = *(const float4*)(I + a * WW + 4);
    s += (u.x + u.y) + (u.z + u.w) + (v.x + v.y) + (v.z + v.w);
  }
  out[t] = s * (1.0f / 64.0f);
}

__global__ void ratio_kernel(const float* __restrict__ mean,
                             const float* __restrict__ source,
                             const float* __restrict__ mask,
                             const float* __restrict__ scal,
                             float* __restrict__ ratio) {
  int t = blockIdx.x * 256 + threadIdx.x;
  float shift = scal[1];
  ratio[t] = (mask[t] < 0.5f) ? 1.0f
                              : (source[t] + shift) / (mean[t] + 1e-8f);
}

// 4 consecutive pixels never straddle an 8-px block: one ratio per quad.
__global__ void apply_ratio4_kernel(const float4* __restrict__ imgt,
                                    const float* __restrict__ ratio,
                                    float4* __restrict__ img) {
  int idx = blockIdx.x * 256 + threadIdx.x;    // over NTOT/4
  int p = idx >> 18;                            // PLANE/4 = 2^18
  int pix4 = idx & ((PLANE / 4) - 1);
  int r = pix4 >> 8;                            // (pix4*4) >> 10
  int c = (pix4 & 255) * 4;
  float rt = ratio[(p << 14) + ((r >> 3) << 7) + (c >> 3)];
  float4 v = imgt[idx];
  v.x *= rt; v.y *= rt; v.z *= rt; v.w *= rt;
  img[idx] = v;
}

__global__ void final4_kernel(const float4* __restrict__ img,
                              const float* __restrict__ scal,
                              float4* __restrict__ out) {
  int idx = blockIdx.x * 256 + threadIdx.x;
  float sh = scal[1];
  float4 v = img[idx];
  v.x -= sh; v.y -= sh; v.z -= sh; v.w -= sh;
  out[idx] = v;
}

// ---------------------------------------------------------------------------
extern "C" void kernel_launch(void* const* d_in, const int* in_sizes, int n_in,
                              void* d_out, int out_size, void* d_ws,
                              size_t ws_size, hipStream_t stream) {
  const float* guide     = (const float*)d_in[0];
  const float* source    = (const float*)d_in[1];
  const float* mask_lr   = (const float*)d_in[2];
  const float* y_bicubic = (const float*)d_in[3];
  float* out = (float*)d_out;

  float* ws = (float*)d_ws;
  const size_t N = (size_t)NTOT;
  float* img = ws;
  float* Are = ws + 1 * N;
  float* Aim = ws + 2 * N;
  float* Bre = ws + 3 * N;
  float* Bim = ws + 4 * N;
  float* cv  = ws + 5 * N;
  float* ch  = ws + 6 * N;
  float* ssm = ws + 7 * N;          // 65536
  float* rat = ssm + 65536;         // 65536
  float* scal = rat + 65536;        // [0]=min bits, [1]=shift

  const dim3 b256(256), gN(NTOT / 256), gN4(NTOT / 4 / 256), gSS(65536 / 256);
  const dim3 gFFT(1024), bFFT(128);
  const dim3 gT(32, 32, 4), bT(256);
  const float SC = 1.0f / 32.0f;    // ortho scaling per 1D pass

  init_scalars_kernel<<<1, 1, 0, stream>>>(scal);
  zero4_kernel<<<gN4, b256, 0, stream>>>((float4*)cv);
  zero4_kernel<<<gN4, b256, 0, stream>>>((float4*)ch);
  minreduce_kernel<<<64, 256, 0, stream>>>(source, 65536, (unsigned*)scal);
  shift_kernel<<<1, 1, 0, stream>>>(scal);
  init_img4_kernel<<<gN4, b256, 0, stream>>>((const float4*)y_bicubic, scal,
                                             (float4*)img);

  // ---- Edge coefficients: per feats channel fft2 -> HP mask -> ifft2 ----
  for (int c = 0; c < 4; ++c) {
    const float* base = (c < 3) ? (guide + (size_t)c * PLANE) : img;
    long pstride = (c < 3) ? (long)(3 * PLANE) : (long)PLANE;
    fft1024_lines<<<gFFT, bFFT, 0, stream>>>(base, nullptr, Are, Aim,
                                             pstride, PLANE, -1.0f, SC, 0);
    transpose2_kernel<<<gT, bT, 0, stream>>>(Are, Bre, Aim, Bim);
    fft1024_lines<<<gFFT, bFFT, 0, stream>>>(Bre, Bim, Are, Aim,
                                             PLANE, PLANE, -1.0f, SC, 1);
    fft1024_lines<<<gFFT, bFFT, 0, stream>>>(Are, Aim, Bre, Bim,
                                             PLANE, PLANE, +1.0f, SC, 0);
    transpose2_kernel<<<gT, bT, 0, stream>>>(Bre, Are, Bim, Aim);
    fft1024_lines<<<gFFT, bFFT, 0, stream>>>(Are, Aim, Bre, Bim,
                                             PLANE, PLANE, +1.0f, SC, 0);
    edge_accum_kernel<<<gN, b256, 0, stream>>>(base, pstride, Bre, cv, ch);
  }
  pm_finalize4_kernel<<<gN4, b256, 0, stream>>>((float4*)cv);
  pm_finalize4_kernel<<<gN4, b256, 0, stream>>>((float4*)ch);

  // ---- 8 diffusion steps ----
  for (int it = 0; it < 8; ++it) {
    fft1024_lines<<<gFFT, bFFT, 0, stream>>>(img, nullptr, Are, Aim,
                                             PLANE, PLANE, -1.0f, SC, 0);
    transpose2_kernel<<<gT, bT, 0, stream>>>(Are, Bre, Aim, Bim);
    fft1024_lines<<<gFFT, bFFT, 0, stream>>>(Bre, Bim, Are, Aim,
                                             PLANE, PLANE, -1.0f, SC, 0);
    // inverse pass 1 with fused 3x3 frequency-domain blur on load
    fft1024_lines<<<gFFT, bFFT, 0, stream>>>(Are, Aim, Bre, Bim,
                                             PLANE, PLANE, +1.0f, SC, 2);
    transpose2_kernel<<<gT, bT, 0, stream>>>(Bre, Are, Bim, Aim);
    fft1024_lines<<<gFFT, bFFT, 0, stream>>>(Are, Aim, Bre, Bim,
                                             PLANE, PLANE, +1.0f, SC, 0);
    // Bre = Re(ifft2(blur(fft2(img)))) = I_freq
    diffuse_kernel<<<gN, b256, 0, stream>>>(img, Bre, cv, ch, Are);
    blockmean_kernel<<<gSS, b256, 0, stream>>>(Are, ssm);
    ratio_kernel<<<gSS, b256, 0, stream>>>(ssm, source, mask_lr, scal, rat);
    apply_ratio4_kernel<<<gN4, b256, 0, stream>>>((const float4*)Are, rat,
                                                  (float4*)img);
  }

  final4_kernel<<<gN4, b256, 0, stream>>>((const float4*)img, scal,
                                          (float4*)out);
}